// GraphConvolutional_21345987461239
// MI455X (gfx1250) — compile-verified
//
#include <hip/hip_runtime.h>
#include <hip/hip_bf16.h>

// CDNA5 / gfx1250, wave32. Fused GraphConvolutional forward:
//   S' = normalize(sim-gated adj), out = S' @ (x @ W) + bias
// Big GEMM runs on V_WMMA_F32_16X16X4_F32 (exact fp32 like the reference).

typedef __attribute__((ext_vector_type(2))) float v2f;
typedef __attribute__((ext_vector_type(8))) float v8f;

#define N_NODES 7
#define DDIM    128
#define ODIM    128
#define WPAD    132   // 128 + 4 -> conflict-free ds_load_b64 for B fragments

__global__ __launch_bounds__(256)
void gcn_wmma_kernel(const float* __restrict__ x,     // [B,7,128]
                     const float* __restrict__ adj,   // [7,7]
                     const float* __restrict__ wgt,   // [128,128] (d-major)
                     const float* __restrict__ bias,  // [128]
                     float* __restrict__ out)         // [B,7,128]
{
    __shared__ float wt[ODIM * WPAD];   // transposed weight: wt[o*WPAD + d]

    const int tid = threadIdx.x;

    // ---- cooperative transposed load of weight into LDS ----
    #pragma unroll
    for (int t = 0; t < (DDIM * ODIM) / 256; ++t) {
        int idx = tid + t * 256;          // linear over [d][o]
        int d = idx >> 7;
        int o = idx & 127;
        wt[o * WPAD + d] = wgt[idx];
    }
    __syncthreads();

    const int lane = tid & 31;
    const int wave = tid >> 5;       // 0..7
    const int grp  = lane >> 4;      // half-wave: 0 -> batch b0, 1 -> batch b0+1
    const int lj   = lane & 15;
    const int kc   = grp * 2;        // K sub-column for A/B fragments

    const int b0  = blockIdx.x * 16 + wave * 2;   // 2 batches per wave
    const int myB = b0 + grp;                     // batch owned by this half-wave

    // ================= Phase 1: S' (7x7) for myB, per-lane redundant ============
    // Each lane holds dims [8*lj, 8*lj+8) of all 7 node features.
    float xr[N_NODES][8];
    {
        const float* xb = x + (size_t)myB * N_NODES * DDIM + lj * 8;
        #pragma unroll
        for (int n = 0; n < N_NODES; ++n) {
            float4 lo = *(const float4*)(xb + n * DDIM);
            float4 hi = *(const float4*)(xb + n * DDIM + 4);
            xr[n][0] = lo.x; xr[n][1] = lo.y; xr[n][2] = lo.z; xr[n][3] = lo.w;
            xr[n][4] = hi.x; xr[n][5] = hi.y; xr[n][6] = hi.z; xr[n][7] = hi.w;
        }
    }

    // Gram upper triangle incl. diagonal via 16-lane xor reductions.
    float g[N_NODES][N_NODES];
    #pragma unroll
    for (int i = 0; i < N_NODES; ++i) {
        #pragma unroll
        for (int j = i; j < N_NODES; ++j) {
            float p = 0.f;
            #pragma unroll
            for (int c = 0; c < 8; ++c) p = fmaf(xr[i][c], xr[j][c], p);
            p += __shfl_xor(p, 1, 16);
            p += __shfl_xor(p, 2, 16);
            p += __shfl_xor(p, 4, 16);
            p += __shfl_xor(p, 8, 16);
            g[i][j] = p;   // broadcast to all lanes of the 16-group
        }
    }

    // sim, column L1 norms, diag reset, adj mask -> S' (upper triangular + diag)
    float sim[N_NODES][N_NODES];
    float denom[N_NODES];
    #pragma unroll
    for (int j = 0; j < N_NODES; ++j) denom[j] = 1.f;   // eye contribution
    #pragma unroll
    for (int i = 0; i < N_NODES; ++i) {
        #pragma unroll
        for (int j = i + 1; j < N_NODES; ++j) {
            float a  = adj[i * N_NODES + j];
            float d2 = fmaxf(g[i][i] + g[j][j] - 2.f * g[i][j], 0.f);
            float e  = __expf(-0.5f * sqrtf(d2));
            float s  = (a > 0.f) ? e : 0.f;
            sim[i][j] = s;
            denom[j] += s;
        }
    }
    #pragma unroll
    for (int j = 0; j < N_NODES; ++j) denom[j] = fmaxf(denom[j], 1e-12f);

    float sp[N_NODES][N_NODES];   // new_adj
    #pragma unroll
    for (int i = 0; i < N_NODES; ++i) {
        #pragma unroll
        for (int j = 0; j < N_NODES; ++j) {
            if (i == j)     sp[i][j] = adj[i * N_NODES + i];            // diag = 1 * adj
            else if (i < j) sp[i][j] = (sim[i][j] / denom[j]) * adj[i * N_NODES + j];
            else            sp[i][j] = 0.f;
        }
    }

    // ================= Phase 2: resident A fragments (16x128 tile) ==============
    // A layout (16x4 f32): lane L -> M = L%16, VGPR0/1 = K = 2*(L/16) + {0,1}.
    // Tile rows: 0..6 = b0 nodes, 7 = zero pad, 8..14 = b0+1 nodes, 15 = zero pad.
    v2f A[32];
    {
        const int  m     = lj;
        const bool valid = (m & 7) != 7;
        v2f zero; zero[0] = 0.f; zero[1] = 0.f;
        #pragma unroll
        for (int k = 0; k < 32; ++k) A[k] = zero;
        if (valid) {
            const int ab = b0 + (m >> 3);
            const int an = m & 7;
            const float* ar = x + ((size_t)(ab * N_NODES + an)) * DDIM + kc;
            #pragma unroll
            for (int k = 0; k < 32; ++k)
                A[k] = *(const v2f*)(ar + 4 * k);
        }
    }

    // ================= Phase 3: WMMA GEMM + fused S' contraction ================
    float* ob_base = out + (size_t)myB * N_NODES * DDIM + lj;

    #pragma unroll 1
    for (int n = 0; n < 8; ++n) {
        v8f c = {};
        // B layout (4x16 f32): lane L -> N = 16n + L%16, VGPR0/1 = K = 2*(L/16)+{0,1}
        const float* bb = &wt[(16 * n + lj) * WPAD + kc];
        #pragma unroll
        for (int k = 0; k < 32; ++k) {
            v2f bv = *(const v2f*)(bb + 4 * k);
            c = __builtin_amdgcn_wmma_f32_16x16x4_f32(
                    /*neg_a=*/false, A[k], /*neg_b=*/false, bv,
                    /*c_mod=*/(short)0, c, /*reuse_a=*/false, /*reuse_b=*/false);
        }
        // C layout: VGPR v of half-wave grp holds support[myB][node v][o=16n+lj]
        float bo = bias[16 * n + lj];
        float* ob = ob_base + 16 * n;
        #pragma unroll
        for (int mr = 0; mr < N_NODES; ++mr) {
            float acc = bo;
            #pragma unroll
            for (int j = 0; j < N_NODES; ++j)
                acc = fmaf(sp[mr][j], c[j], acc);
            ob[mr * DDIM] = acc;
        }
    }
}

extern "C" void kernel_launch(void* const* d_in, const int* in_sizes, int n_in,
                              void* d_out, int out_size, void* d_ws, size_t ws_size,
                              hipStream_t stream) {
    const float* x    = (const float*)d_in[0];
    const float* adj  = (const float*)d_in[1];
    const float* wgt  = (const float*)d_in[2];
    const float* bias = (const float*)d_in[3];
    float* out = (float*)d_out;

    const int B = in_sizes[0] / (N_NODES * DDIM);   // 131072
    const int batchesPerBlock = 16;                 // 8 waves x 2 batches
    dim3 grid(B / batchesPerBlock), block(256);
    hipLaunchKernelGGL(gcn_wmma_kernel, grid, block, 0, stream,
                       x, adj, wgt, bias, out);
}